// ECELoss_26740466385000
// MI455X (gfx1250) — compile-verified
//
#include <hip/hip_runtime.h>
#include <hip/hip_bf16.h>
#include <math.h>

// ECE / max-calibration-error / Brier, one pass over 400 MB (HBM floor ~17us).
// Lane-per-row layout, ~10 wave-instructions per 400B row:
//  - each wave stages 32 consecutive rows (12.8 KB contiguous) into LDS via
//    25x global_load_async_to_lds_b128 with literal IOFFSETs, double-buffered
//  - each lane scans its own row from LDS (25x ds_load_b128; 400B row stride
//    tiles all 64 banks -> minimal 2-pass conflicts), no shuffles
//  - s/q accumulated as float2 -> v_pk_add_f32 / v_pk_fma_f32 (VOP3P)
//  - acc via (p[label] == rowmax); binned stats via ds_add_f32, 46 global
//    atomics per block; tiny zero + finalize kernels.

#define N_BINS 15
#define CLS    100               // classes per row (problem-fixed)
#define RPC    32                // rows per chunk == wave32 lanes
#define CHF    (RPC * CLS)       // floats per chunk buffer (3200)
#define WPB    2                 // waves per block
#define TPB    (WPB * 32)
#define NACC   (3 * N_BINS + 1)  // cnt[15] | conf[15] | acc[15] | bri
#define CPW    5                 // target chunks per wave (load balance)

typedef float v2f __attribute__((ext_vector_type(2)));

#if defined(__AMDGCN__) && \
    __has_builtin(__builtin_amdgcn_global_load_async_to_lds_b128) && \
    __has_builtin(__builtin_amdgcn_s_wait_asynccnt)
#define HAVE_ASYNC 1
#else
#define HAVE_ASYNC 0
#endif

#if HAVE_ASYNC
typedef int v4i __attribute__((vector_size(4 * sizeof(int))));
typedef __attribute__((address_space(1))) v4i gv4i;
typedef __attribute__((address_space(3))) v4i lv4i;

// IOFFSET (bytes) is applied to BOTH the global and LDS address (ISA 08 §4.4),
// and must be a literal -> macro, not a loop variable.
#define ASYNC_CP16(ldsp, gp, off)                                     \
  __builtin_amdgcn_global_load_async_to_lds_b128(                     \
      (gv4i*)(void*)const_cast<float*>(gp), (lv4i*)(void*)(ldsp),     \
      (off), 0)

#define STAGE_FULL(lb, gb)                                            \
  do {                                                                \
    ASYNC_CP16(lb, gb, 0);     ASYNC_CP16(lb, gb, 512);               \
    ASYNC_CP16(lb, gb, 1024);  ASYNC_CP16(lb, gb, 1536);              \
    ASYNC_CP16(lb, gb, 2048);  ASYNC_CP16(lb, gb, 2560);              \
    ASYNC_CP16(lb, gb, 3072);  ASYNC_CP16(lb, gb, 3584);              \
    ASYNC_CP16(lb, gb, 4096);  ASYNC_CP16(lb, gb, 4608);              \
    ASYNC_CP16(lb, gb, 5120);  ASYNC_CP16(lb, gb, 5632);              \
    ASYNC_CP16(lb, gb, 6144);  ASYNC_CP16(lb, gb, 6656);              \
    ASYNC_CP16(lb, gb, 7168);  ASYNC_CP16(lb, gb, 7680);              \
    ASYNC_CP16(lb, gb, 8192);  ASYNC_CP16(lb, gb, 8704);              \
    ASYNC_CP16(lb, gb, 9216);  ASYNC_CP16(lb, gb, 9728);              \
    ASYNC_CP16(lb, gb, 10240); ASYNC_CP16(lb, gb, 10752);             \
    ASYNC_CP16(lb, gb, 11264); ASYNC_CP16(lb, gb, 11776);             \
    ASYNC_CP16(lb, gb, 12288);                                        \
  } while (0)

__device__ __forceinline__ void wait_ds0() {
#if __has_builtin(__builtin_amdgcn_s_wait_dscnt)
  __builtin_amdgcn_s_wait_dscnt(0);
#else
  asm volatile("s_wait_dscnt 0" ::: "memory");
#endif
}
#endif  // HAVE_ASYNC

__global__ void ece_zero(float* __restrict__ ws) {
  int t = threadIdx.x;
  if (t < NACC) ws[t] = 0.f;
}

__global__ __launch_bounds__(TPB) void ece_partial(
    const float* __restrict__ probs, const int* __restrict__ labels,
    float* __restrict__ ws, int N) {
  __shared__ __align__(16) float stage[WPB][2][CHF];  // 51.2 KB
  __shared__ float accl[NACC];

  const int tid  = threadIdx.x;
  const int lane = tid & 31;
  const int wv   = tid >> 5;
  if (tid < NACC) accl[tid] = 0.f;
  __syncthreads();

  const int gw      = blockIdx.x * WPB + wv;   // global wave id
  const int stride  = gridDim.x * WPB;         // waves total
  const int nChunks = (N + RPC - 1) / RPC;

  const long maxF = (long)N * CLS - 4;         // last valid 16B-aligned float4

#if HAVE_ASYNC
  // stage chunk c into buffer `buf` (always issues exactly 25 async ops)
  auto stage_chunk = [&](int buf, int c) {
    float* lb = &stage[wv][buf][lane * 4];
    const long base = (long)c * CHF + lane * 4;
    const float* gb = probs + base;
    if ((long)(c + 1) * CHF <= (long)N * CLS) {
      STAGE_FULL(lb, gb);
    } else {  // tail: clamp global address in-bounds, keep exactly 25 issues
#pragma unroll
      for (int i = 0; i < 25; ++i) {
        long o = (long)i * 128;
        if (base + o > maxF) o = maxF - base;
        ASYNC_CP16(lb + i * 128, gb + o, 0);
      }
    }
  };

  if (gw < nChunks) stage_chunk(0, gw);
  if (gw + stride < nChunks) stage_chunk(1, gw + stride);
#endif

  float briAcc = 0.f;

  int it = 0;
  for (int c = gw; c < nChunks; c += stride, ++it) {
    const int buf = it & 1;
    const int r0  = c * RPC;
    const int row = r0 + lane;       // this lane's row
    const bool rv = row < N;
    const int lab = rv ? labels[row] : 0;

#if HAVE_ASYNC
    if (c + stride < nChunks) __builtin_amdgcn_s_wait_asynccnt(25);
    else                      __builtin_amdgcn_s_wait_asynccnt(0);
    asm volatile("" ::: "memory");
    const float* lrow = &stage[wv][buf][lane * CLS];
#else
    const float* lrow;
    {  // fallback: copy rows to LDS through VGPRs (still lane-per-row compute)
      float* lb = &stage[wv][buf][0];
      const float* gb = probs + (long)r0 * CLS;
#pragma unroll
      for (int i = 0; i < 25; ++i) {
        long o = (long)i * 128 + lane * 4;
        long go = (long)r0 * CLS + o;
        float4 t = (go <= maxF) ? *(const float4*)(gb + o)
                                : make_float4(0.f, 0.f, 0.f, 0.f);
        *(float4*)(lb + o) = t;
      }
      __builtin_amdgcn_s_barrier();
      lrow = &stage[wv][buf][lane * CLS];
    }
#endif

    // per-lane scan of one row: packed f32 accumulation (v_pk_add/fma_f32)
    v2f s2 = {0.f, 0.f}, q2 = {0.f, 0.f};
    float mx = -__builtin_inff();
    const float4* vp = reinterpret_cast<const float4*>(lrow);
#pragma unroll
    for (int k = 0; k < CLS / 4; ++k) {
      float4 v = vp[k];                       // ds_load_b128
      v2f a = {v.x, v.y};
      v2f b = {v.z, v.w};
      s2 += a + b;
      q2 += a * a;                            // contracts to v_pk_fma_f32
      q2 += b * b;
      mx = fmaxf(mx, fmaxf(fmaxf(v.x, v.y), fmaxf(v.z, v.w)));
    }
    const float s = s2.x + s2.y;
    const float q = q2.x + q2.y;

    if (rv) {
      const float pl   = lrow[lab];           // ds_load_b32
      const float inv  = 1.f / s;
      const float conf = mx * inv;
      const float accv = (pl == mx) ? 1.f : 0.f;
      briAcc += (q * inv) * inv - 2.f * pl * inv + 1.f;
      int bin = (int)ceilf(conf * (float)N_BINS) - 1;
      bin = bin < 0 ? 0 : (bin > N_BINS - 1 ? N_BINS - 1 : bin);
      atomicAdd(&accl[bin], 1.f);
      atomicAdd(&accl[N_BINS + bin], conf);
      atomicAdd(&accl[2 * N_BINS + bin], accv);
    }

#if HAVE_ASYNC
    wait_ds0();                         // LDS reads done before overwrite
    asm volatile("" ::: "memory");
    const int cnn = c + 2 * stride;
    if (cnn < nChunks) stage_chunk(buf, cnn);
    {  // L2 prefetch one chunk beyond the pipeline
      const int cp = c + 3 * stride;
      if (cp < nChunks)
        __builtin_prefetch(probs + (long)cp * CHF + lane * CLS, 0, 1);
    }
#endif
  }

  // reduce per-lane Brier partials across the wave, then to block, then global
#pragma unroll
  for (int off = 16; off > 0; off >>= 1) briAcc += __shfl_xor(briAcc, off, 32);
  if (lane == 0) atomicAdd(&accl[3 * N_BINS], briAcc);
  __syncthreads();
  if (tid < NACC) atomicAdd(&ws[tid], accl[tid]);
}

__global__ void ece_final(const float* __restrict__ ws, float* __restrict__ out,
                          float Nf) {
  if (threadIdx.x == 0 && blockIdx.x == 0) {
    float ece = 0.f, mxg = 0.f;
    for (int b = 0; b < N_BINS; ++b) {
      float c = ws[b];
      if (c > 0.f) {
        float gap = fabsf(ws[N_BINS + b] / c - ws[2 * N_BINS + b] / c);
        ece += gap * (c / Nf);
        if (gap > mxg) mxg = gap;
      }
    }
    out[0] = ece;
    out[1] = mxg;
    out[2] = ws[3 * N_BINS] / Nf;
  }
}

extern "C" void kernel_launch(void* const* d_in, const int* in_sizes, int n_in,
                              void* d_out, int out_size, void* d_ws, size_t ws_size,
                              hipStream_t stream) {
  const float* probs  = (const float*)d_in[0];
  const int*   labels = (const int*)d_in[1];
  float*       ws     = (float*)d_ws;
  float*       out    = (float*)d_out;

  const int N = in_sizes[1];   // rows; CLS (=100) is problem-fixed

  ece_zero<<<1, 64, 0, stream>>>(ws);

  const int nChunks = (N + RPC - 1) / RPC;
  // exactly CPW chunks per wave when possible (perfect balance for N=1e6)
  int blocks = (nChunks + WPB * CPW - 1) / (WPB * CPW);
  if (blocks > nChunks) blocks = nChunks;
  if (blocks < 1) blocks = 1;
  ece_partial<<<blocks, TPB, 0, stream>>>(probs, labels, ws, N);

  ece_final<<<1, 32, 0, stream>>>(ws, out, (float)N);
}